// RotatE_8924942041236
// MI455X (gfx1250) — compile-verified
//
#include <hip/hip_runtime.h>
#include <hip/hip_bf16.h>

#define EMB        512
#define ROWS       8            // entity rows per LDS tile (one per wave)
#define NQ         8            // batch of queries (fixed in reference)
#define GAMMA_F    12.0f
// phase scale = pi / ((GAMMA+2)/EMB)
#define PHASE_SCALE 114.85946045293465f   // 3.141592653589793 / 0.02734375

#if __has_builtin(__builtin_amdgcn_global_load_async_to_lds_b128)
#define HAVE_ASYNC_LDS 1
#else
#define HAVE_ASYNC_LDS 0
#endif

typedef int v4i __attribute__((ext_vector_type(4)));
typedef __attribute__((address_space(1))) v4i gv4i;   // global 16B chunk
typedef __attribute__((address_space(3))) v4i lv4i;   // LDS 16B chunk
typedef float v2f __attribute__((ext_vector_type(2)));

__device__ __forceinline__ void wait_async_all() {
#if __has_builtin(__builtin_amdgcn_s_wait_asynccnt)
    __builtin_amdgcn_s_wait_asynccnt(0);
#elif HAVE_ASYNC_LDS
    asm volatile("s_wait_asynccnt 0" ::: "memory");
#endif
}

// ---- CDNA5 VOP3P packed-f32 helpers (2 lanes of f32 per instruction) ------
__device__ __forceinline__ v2f pk_sub(v2f a, v2f b) {   // a - b per half
    v2f d;
    asm("v_pk_add_f32 %0, %1, %2 neg_lo:[0,1] neg_hi:[0,1]"
        : "=v"(d) : "v"(a), "v"(b));
    return d;
}
__device__ __forceinline__ v2f pk_mul(v2f a, v2f b) {
    v2f d;
    asm("v_pk_mul_f32 %0, %1, %2" : "=v"(d) : "v"(a), "v"(b));
    return d;
}
__device__ __forceinline__ v2f pk_fma(v2f a, v2f b, v2f c) {  // a*b + c
    v2f d;
    asm("v_pk_fma_f32 %0, %1, %2, %3" : "=v"(d) : "v"(a), "v"(b), "v"(c));
    return d;
}
__device__ __forceinline__ void pk_acc(v2f& acc, v2f x) {     // acc += x
    asm("v_pk_add_f32 %0, %1, %0" : "+v"(acc) : "v"(x));
}

// ---------------------------------------------------------------------------
// Kernel 1: rotate the 8 query embeddings by their relation phases.
// Output layout matches entity rows: qout[b*1024 + k] = re_e, [+512] = im_e.
// ---------------------------------------------------------------------------
__global__ __launch_bounds__(256) void rotate_queries(
    const int* __restrict__ all_h, const int* __restrict__ all_r,
    const float* __restrict__ eemb, const float* __restrict__ remb,
    float* __restrict__ qout) {
    const int b = blockIdx.x;
    const int h = all_h[b];
    const int r = all_r[b];
    const float* hrow = eemb + (size_t)h * (2 * EMB);
    const float* rrow = remb + (size_t)r * EMB;
    for (int k = threadIdx.x; k < EMB; k += blockDim.x) {
        float re_h = hrow[k];
        float im_h = hrow[EMB + k];
        float ph   = rrow[k] * PHASE_SCALE;
        float c = cosf(ph);
        float s = sinf(ph);
        qout[b * (2 * EMB) + k]       = re_h * c - im_h * s;   // re_e
        qout[b * (2 * EMB) + EMB + k] = re_h * s + im_h * c;   // im_e
    }
}

// ---------------------------------------------------------------------------
// Issue async copy of one tile (ROWS entity rows) into an LDS buffer.
// Wave w copies row (tile*ROWS + w): 4 KB = 8 x (32 lanes x b128).
// ---------------------------------------------------------------------------
__device__ __forceinline__ void issue_tile(const float* __restrict__ eemb,
                                           int tile, int nEnt,
                                           float* dstbase, int wave, int lane) {
    const int row = tile * ROWS + wave;
    if (row >= nEnt) return;
    const char* src = (const char*)(eemb + (size_t)row * (2 * EMB));
    char*       dst = (char*)(dstbase + wave * (2 * EMB));
#if HAVE_ASYNC_LDS
#pragma unroll
    for (int j = 0; j < 8; ++j) {
        const int off = j * 512 + lane * 16;
        __builtin_amdgcn_global_load_async_to_lds_b128(
            (gv4i*)(src + off), (lv4i*)(dst + off), 0, 0);
    }
#else
    const float4* s4 = (const float4*)src;
    float4*       d4 = (float4*)dst;
#pragma unroll
    for (int j = 0; j < 8; ++j) d4[j * 32 + lane] = s4[j * 32 + lane];
#endif
}

// ---------------------------------------------------------------------------
// Kernel 2: distances to all entities. 8 waves/block, one entity row per wave.
// Double-buffered async LDS tiles; packed-f32 VOP3P math, 4 dims/lane/step.
// ---------------------------------------------------------------------------
__global__ __launch_bounds__(256) void rotate_scores(
    const float* __restrict__ eemb, const float* __restrict__ qbuf,
    float* __restrict__ out, int nEnt, int numTiles) {
    __shared__ __align__(16) float qlds[NQ * 2 * EMB];        // 32 KB
    __shared__ __align__(16) float tilebuf[2][ROWS * 2 * EMB]; // 2 x 32 KB

    const int tid  = threadIdx.x;
    const int wave = tid >> 5;
    const int lane = tid & 31;

    // Stage rotated queries once (broadcast-shared by every tile).
    {
        const float4* qs = (const float4*)qbuf;
        float4*       qd = (float4*)qlds;
#pragma unroll
        for (int i = 0; i < (NQ * 2 * EMB) / (4 * 256); ++i)
            qd[i * 256 + tid] = qs[i * 256 + tid];
    }
    __syncthreads();

    int t = blockIdx.x;
    const int stride = gridDim.x;
    int cur = 0;
    if (t < numTiles) issue_tile(eemb, t, nEnt, tilebuf[0], wave, lane);

    for (; t < numTiles; t += stride) {
        wait_async_all();      // this wave's async loads into tilebuf[cur] done
        __syncthreads();       // => all waves' loads done, tile visible

        const int nt = t + stride;
        if (nt < numTiles) issue_tile(eemb, nt, nEnt, tilebuf[cur ^ 1], wave, lane);

        const int row = t * ROWS + wave;
        if (row < nEnt) {
            const float* trow = &tilebuf[cur][wave * (2 * EMB)];
            v2f acc[NQ];
#pragma unroll
            for (int b = 0; b < NQ; ++b) acc[b] = (v2f){0.0f, 0.0f};

#pragma unroll
            for (int j = 0; j < EMB / 128; ++j) {          // 4 chunks
                const int k4 = j * 128 + lane * 4;         // 4 dims per lane
                const float4 tre4 = *(const float4*)&trow[k4];         // b128
                const float4 tim4 = *(const float4*)&trow[EMB + k4];   // b128
                const v2f tre0 = (v2f){tre4.x, tre4.y}, tre1 = (v2f){tre4.z, tre4.w};
                const v2f tim0 = (v2f){tim4.x, tim4.y}, tim1 = (v2f){tim4.z, tim4.w};
#pragma unroll
                for (int b = 0; b < NQ; ++b) {
                    const float* qrow = &qlds[b * (2 * EMB)];
                    const float4 qre4 = *(const float4*)&qrow[k4];       // b128
                    const float4 qim4 = *(const float4*)&qrow[EMB + k4]; // b128
                    // pair 0 (dims k4, k4+1)
                    v2f dr = pk_sub((v2f){qre4.x, qre4.y}, tre0);
                    v2f di = pk_sub((v2f){qim4.x, qim4.y}, tim0);
                    v2f s2 = pk_mul(dr, dr);
                    s2 = pk_fma(di, di, s2);
                    v2f rt;
                    rt.x = __builtin_amdgcn_sqrtf(s2.x);
                    rt.y = __builtin_amdgcn_sqrtf(s2.y);
                    pk_acc(acc[b], rt);
                    // pair 1 (dims k4+2, k4+3)
                    dr = pk_sub((v2f){qre4.z, qre4.w}, tre1);
                    di = pk_sub((v2f){qim4.z, qim4.w}, tim1);
                    s2 = pk_mul(dr, dr);
                    s2 = pk_fma(di, di, s2);
                    rt.x = __builtin_amdgcn_sqrtf(s2.x);
                    rt.y = __builtin_amdgcn_sqrtf(s2.y);
                    pk_acc(acc[b], rt);
                }
            }
            // wave32 butterfly reduction per query, lane 0 writes result
#pragma unroll
            for (int b = 0; b < NQ; ++b) {
                float v = acc[b].x + acc[b].y;
#pragma unroll
                for (int off = 16; off > 0; off >>= 1) v += __shfl_xor(v, off, 32);
                if (lane == 0) out[(size_t)b * nEnt + row] = GAMMA_F - v;
            }
        }
        __syncthreads();       // tile consumed before it can be refilled
        cur ^= 1;
    }
}

extern "C" void kernel_launch(void* const* d_in, const int* in_sizes, int n_in,
                              void* d_out, int out_size, void* d_ws, size_t ws_size,
                              hipStream_t stream) {
    const int*   all_h = (const int*)d_in[0];
    const int*   all_r = (const int*)d_in[1];
    const float* eemb  = (const float*)d_in[2];
    const float* remb  = (const float*)d_in[3];
    float*       out   = (float*)d_out;
    float*       qbuf  = (float*)d_ws;             // 8*1024 floats = 32 KB

    const int nEnt = in_sizes[2] / (2 * EMB);      // 30000

    rotate_queries<<<NQ, 256, 0, stream>>>(all_h, all_r, eemb, remb, qbuf);

    const int numTiles = (nEnt + ROWS - 1) / ROWS; // 3750
    int grid = (numTiles + 2) / 3;                 // ~3 tiles per block for
    if (grid < 1) grid = 1;                        // double-buffer pipelining
    rotate_scores<<<grid, 256, 0, stream>>>(eemb, qbuf, out, nEnt, numTiles);
}